// skipConnection_35527969472878
// MI455X (gfx1250) — compile-verified
//
#include <hip/hip_runtime.h>

// out[i] = in[i,0]*w + in[i,1]*(1-w)   (N = 2^24 rows, f32)
// Pure streaming kernel: ~201 MB total traffic, roofline ~8.6 us @ 23.3 TB/s.
// Strategy: widest possible VMEM (b128 loads/stores), non-temporal cache
// policy (single-touch data), zero tail code (N divides evenly).

typedef float f32x4 __attribute__((ext_vector_type(4)));

#define THREADS_PER_BLOCK 256
#define ROWS_PER_THREAD   8   // 4x b128 loads, 2x b128 stores per thread

__global__ __launch_bounds__(THREADS_PER_BLOCK) void skip_connection_kernel(
    const f32x4* __restrict__ in,    // N/2 float4s: (a0,b0,a1,b1)...
    const float* __restrict__ wptr,  // single scalar weight
    f32x4* __restrict__ out)         // N/4 float4s
{
    const float w   = wptr[0];        // uniform -> s_load + broadcast
    const float omw = 1.0f - w;

    const long long t  = (long long)blockIdx.x * THREADS_PER_BLOCK + threadIdx.x;
    const long long ib = t * 4;       // input float4 index (8 rows = 4 float4)
    const long long ob = t * 2;       // output float4 index

    // 4 outstanding 128-bit non-temporal loads (64 B/thread, 2 KB/wave clause)
    f32x4 v0 = __builtin_nontemporal_load(&in[ib + 0]);
    f32x4 v1 = __builtin_nontemporal_load(&in[ib + 1]);
    f32x4 v2 = __builtin_nontemporal_load(&in[ib + 2]);
    f32x4 v3 = __builtin_nontemporal_load(&in[ib + 3]);

    f32x4 o0, o1;
    o0.x = fmaf(v0.x, w, v0.y * omw);
    o0.y = fmaf(v0.z, w, v0.w * omw);
    o0.z = fmaf(v1.x, w, v1.y * omw);
    o0.w = fmaf(v1.z, w, v1.w * omw);
    o1.x = fmaf(v2.x, w, v2.y * omw);
    o1.y = fmaf(v2.z, w, v2.w * omw);
    o1.z = fmaf(v3.x, w, v3.y * omw);
    o1.w = fmaf(v3.z, w, v3.w * omw);

    __builtin_nontemporal_store(o0, &out[ob + 0]);
    __builtin_nontemporal_store(o1, &out[ob + 1]);
}

extern "C" void kernel_launch(void* const* d_in, const int* in_sizes, int n_in,
                              void* d_out, int out_size, void* d_ws, size_t ws_size,
                              hipStream_t stream) {
    (void)n_in; (void)d_ws; (void)ws_size;

    const float* input  = (const float*)d_in[0];  // [N,2] f32
    const float* weight = (const float*)d_in[1];  // [1,1] f32
    float*       output = (float*)d_out;          // [N,1] f32

    const long long n_rows  = out_size;                       // 16,777,216
    const long long threads = n_rows / ROWS_PER_THREAD;       // 2,097,152
    const int       blocks  = (int)(threads / THREADS_PER_BLOCK); // 8192

    skip_connection_kernel<<<blocks, THREADS_PER_BLOCK, 0, stream>>>(
        (const f32x4*)input, weight, (f32x4*)output);
}